// RoIAlign_9474697855111
// MI455X (gfx1250) — compile-verified
//
#include <hip/hip_runtime.h>
#include <math.h>

typedef __attribute__((ext_vector_type(2))) float v2f;
typedef __attribute__((ext_vector_type(4))) float v4f;
typedef __attribute__((ext_vector_type(8))) float v8f;

#define C_CH 256

// One workgroup per (roi, output_row i). 256 threads = 8 waves (wave32).
// Out row (7 x 256) = W(7x28) x P(28x256) via V_WMMA_F32_16X16X4_F32:
//   M=16 (7 output cols padded), K=28 (7 steps of 4), N=16 channels/tile.
__global__ __launch_bounds__(256) void roi_align_wmma_kernel(
    const float* __restrict__ p2, const float* __restrict__ p3,
    const float* __restrict__ p4, const float* __restrict__ p5,
    const float* __restrict__ rois, float* __restrict__ out, int nroi)
{
    // P: 28 gathered feature rows x 256 channels. Row stride 264 floats:
    // keeps rows 16B aligned for b128 LDS stores and avoids bank conflicts
    // between lane l and l+16 in the B-fragment reads.
    __shared__ float Plds[28][264];
    __shared__ float Wlds[16][28];

    const int blk = blockIdx.x;
    const int roi = blk / 7;
    const int i   = blk % 7;
    const int t   = threadIdx.x;

    // ---- RoI parameters (uniform across the block) ----
    const float bidx_f = rois[roi * 5 + 0];
    const float rx1    = rois[roi * 5 + 1];
    const float ry1    = rois[roi * 5 + 2];
    const float rx2    = rois[roi * 5 + 3];
    const float ry2    = rois[roi * 5 + 4];
    const int   b      = (int)bidx_f;

    const float roi_w = rx2 - rx1;
    const float roi_h = ry2 - ry1;
    // lvl = 4 + log2(sqrt(h*w)/224); jnp.round = half-to-even => rintf
    const float lvl = 4.0f + log2f(sqrtf(roi_h * roi_w) / 224.0f);
    int level = (int)rintf(lvl);
    level = level < 2 ? 2 : (level > 5 ? 5 : level);

    const float* fmap;
    int H;
    float inv_stride;
    switch (level) {
        case 2:  fmap = p2; H = 256; inv_stride = 1.0f / 4.0f;  break;
        case 3:  fmap = p3; H = 128; inv_stride = 1.0f / 8.0f;  break;
        case 4:  fmap = p4; H = 64;  inv_stride = 1.0f / 16.0f; break;
        default: fmap = p5; H = 32;  inv_stride = 1.0f / 32.0f; break;
    }
    const int   W   = H;
    const float Hm1 = (float)(H - 1);
    const float Wm1 = Hm1;

    // Reference quirk: boxes = [x1,y1,x2,y2]*(1/stride) consumed as [y1,x1,y2,x2]
    const float cy1 = rx1 * inv_stride;
    const float cx1 = ry1 * inv_stride;
    const float cy2 = rx2 * inv_stride;
    const float cx2 = ry2 * inv_stride;

    const float h_scale = (cy2 - cy1) * Hm1 / 6.0f;
    const float w_scale = (cx2 - cx1) * Wm1 / 6.0f;

    // y sampling for this output row
    const float in_y    = cy1 * Hm1 + (float)i * h_scale;
    const bool  valid_y = (in_y >= 0.0f) && (in_y <= Hm1);
    const float fy      = floorf(in_y);
    const float ly      = in_y - fy;
    const int   y0      = (int)fminf(fmaxf(fy, 0.0f), Hm1);
    const int   yc      = (int)fminf(fmaxf(ceilf(in_y), 0.0f), Hm1);

    // ---- Build W(16x28) in LDS. Row m = output col j; cols s: 0..13 use
    // feature row y0 (weight 1-ly), 14..27 use yc (weight ly). Within each
    // half, s%14 == 2j -> (1-lx_j), 2j+1 -> lx_j. Validity zeroes the row. ----
    for (int w = t; w < 16 * 28; w += 256) {
        const int m   = w / 28;
        const int s   = w % 28;
        float     val = 0.0f;
        if (m < 7 && valid_y) {
            const int s14 = s % 14;
            const int j   = s14 >> 1;
            if (j == m) {
                const float in_x    = cx1 * Wm1 + (float)j * w_scale;
                const bool  valid_x = (in_x >= 0.0f) && (in_x <= Wm1);
                if (valid_x) {
                    const float lx = in_x - floorf(in_x);
                    const float wy = (s < 14) ? (1.0f - ly) : ly;
                    const float wx = (s14 & 1) ? lx : (1.0f - lx);
                    val = wy * wx;
                }
            }
        }
        Wlds[m][s] = val;
    }

    // ---- Gather P(28 x 256) into LDS. Per iter: 4 P-rows, 64 threads/row,
    // float4 (global_load_b128) along channels -> fully coalesced 1KB/row. ----
    const int pr_base = t >> 6;        // 0..3
    const int cpos    = (t & 63) * 4;  // channel offset, float4 granularity
#pragma unroll
    for (int iter = 0; iter < 7; ++iter) {
        const int pr   = iter * 4 + pr_base;          // 0..27
        const int s    = (pr < 14) ? pr : pr - 14;
        const int yrow = (pr < 14) ? y0 : yc;
        const int j    = s >> 1;
        const float in_x  = cx1 * Wm1 + (float)j * w_scale;
        const float col_f = (s & 1) ? ceilf(in_x) : floorf(in_x);
        const int   col   = (int)fminf(fmaxf(col_f, 0.0f), Wm1);
        const float* src =
            fmap + (((size_t)b * H + yrow) * W + col) * C_CH + cpos;
        *(v4f*)&Plds[pr][cpos] = *(const v4f*)src;
    }
    __syncthreads();

    // ---- WMMA: each wave owns channel tiles nt0=wave, nt1=wave+8. ----
    const int lane = t & 31;
    const int wave = t >> 5;
    const int half = lane >> 4;   // lane half selects K offset +2 / M offset +8
    const int lm   = lane & 15;
    const int nt0  = wave;
    const int nt1  = wave + 8;

    v8f acc0 = {};
    v8f acc1 = {};
#pragma unroll
    for (int k = 0; k < 7; ++k) {
        const int ka = 4 * k + 2 * half;
        // A(16x4) frag: VGPR0 <-> K=ka, VGPR1 <-> K=ka+1 (ISA 7.12.2)
        v2f a;
        a.x = Wlds[lm][ka];
        a.y = Wlds[lm][ka + 1];
        // B(4x16) frags: VGPR0 <-> row K=ka, VGPR1 <-> row K=ka+1
        v2f b0, b1;
        b0.x = Plds[ka][nt0 * 16 + lm];
        b0.y = Plds[ka + 1][nt0 * 16 + lm];
        b1.x = Plds[ka][nt1 * 16 + lm];
        b1.y = Plds[ka + 1][nt1 * 16 + lm];
        acc0 = __builtin_amdgcn_wmma_f32_16x16x4_f32(
            false, a, false, b0, (short)0, acc0, false, false);
        acc1 = __builtin_amdgcn_wmma_f32_16x16x4_f32(
            false, a, false, b1, (short)0, acc1, false, false);
    }

    // ---- Store: D VGPR v holds M = v + 8*half. Only M=0..6 are live, so
    // lanes 0-15 store rows 0..6 (lane half 1 holds discarded pad rows). ----
    if (half == 0) {
        const size_t base = ((size_t)(roi * 7 + i) * 7) * C_CH;
#pragma unroll
        for (int v = 0; v < 7; ++v) {
            out[base + (size_t)v * C_CH + nt0 * 16 + lm] = acc0[v];
            out[base + (size_t)v * C_CH + nt1 * 16 + lm] = acc1[v];
        }
    }
}

extern "C" void kernel_launch(void* const* d_in, const int* in_sizes, int n_in,
                              void* d_out, int out_size, void* d_ws,
                              size_t ws_size, hipStream_t stream)
{
    const float* p2   = (const float*)d_in[0];
    const float* p3   = (const float*)d_in[1];
    const float* p4   = (const float*)d_in[2];
    const float* p5   = (const float*)d_in[3];
    const float* rois = (const float*)d_in[4];
    float*       out  = (float*)d_out;

    const int nroi = in_sizes[4] / 5;

    roi_align_wmma_kernel<<<dim3(nroi * 7), dim3(256), 0, stream>>>(
        p2, p3, p4, p5, rois, out, nroi);
}